// PointDeconv_80272938762655
// MI455X (gfx1250) — compile-verified
//
#include <hip/hip_runtime.h>
#include <hip/hip_bf16.h>

// ---------------- problem constants (from reference) ----------------
constexpr int kB   = 4;
constexpr int kNIN = 4096;
constexpr int kM   = 16384;
constexpr int kCIN = 256;
constexpr int kCO  = 128;
constexpr float kEps = 1e-5f;

typedef __attribute__((ext_vector_type(2))) float v2f;
typedef __attribute__((ext_vector_type(8))) float v8f;

// TDM descriptor vector types (probe-verified arities in CDNA5_HIP.md)
typedef __attribute__((ext_vector_type(4))) unsigned tdm_v4u;
typedef __attribute__((ext_vector_type(8))) int      tdm_v8i;
typedef __attribute__((ext_vector_type(4))) int      tdm_v4i;

#define AS_LDS __attribute__((address_space(3)))
#define AS_GLB __attribute__((address_space(1)))

// async-LDS copy builtin takes (int4 AS1*, int4 AS3*, imm, imm)
__device__ inline AS_GLB tdm_v4i* glb_v4(const void* p) {
    return (AS_GLB tdm_v4i*)(uintptr_t)p;
}
__device__ inline AS_LDS tdm_v4i* lds_v4(void* p) {
    // low 32 bits of a flat LDS address are the wave-relative LDS offset
    return (AS_LDS tdm_v4i*)(unsigned)(uintptr_t)p;
}

// ---------------- K0: fold BN into (alpha, gamma) per channel -------
// y = g*(acc + b - rm)/sqrt(rv+eps) + beta  ==  alpha*acc + gamma
__global__ void fold_bn_kernel(const float* b1, const float* g1, const float* be1,
                               const float* rm1, const float* rv1,
                               const float* b2, const float* g2, const float* be2,
                               const float* rm2, const float* rv2,
                               const float* b3, const float* g3, const float* be3,
                               const float* rm3, const float* rv3,
                               float* pars) {
    int c = threadIdx.x;
    if (c >= kCO) return;
    {
        float a = g1[c] * __frsqrt_rn(rv1[c] + kEps);
        pars[0 * 256 + c]       = a;
        pars[0 * 256 + 128 + c] = a * (b1[c] - rm1[c]) + be1[c];
    }
    {
        float a = g2[c] * __frsqrt_rn(rv2[c] + kEps);
        pars[1 * 256 + c]       = a;
        pars[1 * 256 + 128 + c] = a * (b2[c] - rm2[c]) + be2[c];
    }
    {
        float a = g3[c] * __frsqrt_rn(rv3[c] + kEps);
        pars[2 * 256 + c]       = a;
        pars[2 * 256 + 128 + c] = a * (b3[c] - rm3[c]) + be3[c];
    }
}

// ---------------- K1: fused 3-layer pointwise MLP on WMMA f32 -------
// One block: 16 points (n-tile) x all 128 out channels. 8 waves, wave w owns
// channel tile [16w, 16w+16). Activations exchanged via LDS between layers.
//
// V_WMMA_F32_16X16X4_F32 lane layout (wave32):
//   A (16x4):  a.x = A[lane&15][2*(lane>>4)],   a.y = A[lane&15][2*(lane>>4)+1]
//   B (4x16):  b.x = B[2*(lane>>4)][lane&15],   b.y = B[2*(lane>>4)+1][lane&15]
//   C/D (16x16): c[v] = D[v + 8*(lane>>4)][lane&15]
__device__ inline v8f gemm_tile_f32(const float* __restrict__ W, int ldw,
                                    const float* __restrict__ Xlds, // [K][16]
                                    int K, int co0, int lane) {
    const int nrow = lane & 15;
    const int kq   = (lane >> 4) << 1; // 0 or 2
    v8f acc = {};
    const float* wrow = W + (size_t)(co0 + nrow) * ldw + kq;
    #pragma unroll 4
    for (int k0 = 0; k0 < K; k0 += 4) {
        v2f a, bb;
        a.x  = wrow[k0 + 0];
        a.y  = wrow[k0 + 1];
        bb.x = Xlds[(k0 + kq) * 16 + nrow];
        bb.y = Xlds[(k0 + kq + 1) * 16 + nrow];
        acc = __builtin_amdgcn_wmma_f32_16x16x4_f32(
            /*neg_a=*/false, a, /*neg_b=*/false, bb,
            /*c_mod=*/(short)0, acc, /*reuse_a=*/false, /*reuse_b=*/false);
    }
    return acc;
}

__device__ inline void bn_relu_to_lds(v8f acc, const float* alpha, const float* gamma,
                                      float* dst, int co0, int lane) {
    const int n    = lane & 15;
    const int half = (lane >> 4) << 3; // 0 or 8
    #pragma unroll
    for (int v = 0; v < 8; ++v) {
        int co  = co0 + half + v;
        float y = fmaxf(fmaf(acc[v], alpha[co], gamma[co]), 0.0f);
        dst[co * 16 + n] = y;
    }
}

__global__ __launch_bounds__(256) void mlp_fused_kernel(
    const float* __restrict__ rgb,  // (B, CIN, NIN)
    const float* __restrict__ w1,   // (CO, CIN)
    const float* __restrict__ w2,   // (CO, CO)
    const float* __restrict__ w3,   // (CO, CO)
    const float* __restrict__ pars, // 3 x [alpha(128), gamma(128)]
    float* __restrict__ featT)      // (B, NIN, CO)
{
    __shared__ float xT[kCIN * 16];   // 16 KB: x[k][n]
    __shared__ float actA[kCO * 16];  //  8 KB
    __shared__ float actB[kCO * 16];  //  8 KB

    const int bid  = blockIdx.x;            // 0 .. B*(NIN/16)-1
    const int b    = bid >> 8;              // NIN/16 = 256 tiles per batch
    const int n0   = (bid & 255) << 4;
    const int t    = threadIdx.x;
    const int lane = t & 31;
    const int wv   = t >> 5;                // wave id 0..7 -> channel tile
    const int co0  = wv << 4;

#if __has_builtin(__builtin_amdgcn_tensor_load_to_lds) && \
    __has_builtin(__builtin_amdgcn_s_wait_tensorcnt)
    // ---- TDM: one 2D tensor tile (16 cols x 256 rows, row stride NIN) ----
    if (wv == 0) {
        unsigned lds_off = (unsigned)(uintptr_t)(void*)xT;
        unsigned long long ga =
            (unsigned long long)(uintptr_t)(rgb + (size_t)b * kCIN * kNIN + n0);
        tdm_v4u g0;
        g0[0] = 1u;                                   // count=1, user mode
        g0[1] = lds_off;                              // lds_addr
        g0[2] = (unsigned)(ga & 0xFFFFFFFFull);       // global_addr[31:0]
        g0[3] = (unsigned)((ga >> 32) & 0x01FFFFFFull) | (2u << 30); // type=2
        tdm_v8i g1;
        g1[0] = (int)(2u << 16);                      // data_size = 4 bytes
        g1[1] = (int)((kNIN & 0xFFFF) << 16);         // tensor_dim0[15:0]
        g1[2] = (int)(((unsigned)kNIN >> 16) | ((kCIN & 0xFFFF) << 16)); // td0 hi | td1 lo
        g1[3] = (int)(((unsigned)kCIN >> 16) | (16u << 16));             // td1 hi | tile_dim0
        g1[4] = (int)kCIN;                            // tile_dim1 (tile_dim2 = 0)
        g1[5] = (int)kNIN;                            // tensor_dim0_stride[31:0]
        g1[6] = 0;                                    // stride hi | dim1_stride lo
        g1[7] = 0;
        tdm_v4i gz = {0, 0, 0, 0};
#if __clang_major__ >= 23
        tdm_v8i gz8 = {0, 0, 0, 0, 0, 0, 0, 0};
        __builtin_amdgcn_tensor_load_to_lds(g0, g1, gz, gz, gz8, 0);
#else
        __builtin_amdgcn_tensor_load_to_lds(g0, g1, gz, gz, 0);
#endif
        __builtin_amdgcn_s_wait_tensorcnt(0);
    }
#else
    // ---- fallback: manual staged copy ----
    #pragma unroll
    for (int p = 0; p < 16; ++p) {
        int ci = (p << 4) + (t >> 4);
        int n  = t & 15;
        xT[ci * 16 + n] = rgb[((size_t)b * kCIN + ci) * kNIN + n0 + n];
    }
#endif
    __syncthreads();

    // layer 1: K = 256
    {
        v8f acc = gemm_tile_f32(w1, kCIN, xT, kCIN, co0, lane);
        bn_relu_to_lds(acc, pars + 0 * 256, pars + 0 * 256 + 128, actA, co0, lane);
    }
    __syncthreads();

    // layer 2: K = 128
    {
        v8f acc = gemm_tile_f32(w2, kCO, actA, kCO, co0, lane);
        bn_relu_to_lds(acc, pars + 1 * 256, pars + 1 * 256 + 128, actB, co0, lane);
    }
    __syncthreads();

    // layer 3: K = 128, write featT (B, NIN, CO) directly
    {
        v8f acc = gemm_tile_f32(w3, kCO, actB, kCO, co0, lane);
        const float* alpha = pars + 2 * 256;
        const float* gamma = pars + 2 * 256 + 128;
        const int n    = lane & 15;
        const int half = (lane >> 4) << 3;
        #pragma unroll
        for (int v = 0; v < 8; ++v) {
            int co  = co0 + half + v;
            float y = fmaxf(fmaf(acc[v], alpha[co], gamma[co]), 0.0f);
            featT[((size_t)b * kNIN + n0 + n) * kCO + co] = y;
        }
    }
}

// ---------------- K2: streamed top-3 NN + weighted interpolation ----
// Block: 256 output points. xyzin async-staged to LDS, augmented with |p|^2;
// every lane keeps its own sorted top-3 (d, idx). Then interpolate 3-NN
// features via LDS transpose so global stores are coalesced along M.
__global__ __launch_bounds__(256) void interp_kernel(
    const float* __restrict__ xyzin,   // (B, NIN, 3)
    const float* __restrict__ xyzout,  // (B, M, 3)
    const float* __restrict__ featT,   // (B, NIN, CO)
    float* __restrict__ out)           // (B, CO, M)
{
    __shared__ float  smRaw[1024 * 3];        // 12 KB raw xyz chunk
    __shared__ float4 smPts[1024];            // 16 KB (x,y,z,|p|^2)
    __shared__ float  smOutT[kCO * 65];       // ~33 KB, padded columns
    __shared__ int    smIdx[256 * 3];
    __shared__ float  smW[256 * 3];

    const int b  = blockIdx.x >> 6;           // M/256 = 64 blocks per batch
    const int m0 = (blockIdx.x & 63) << 8;
    const int t  = threadIdx.x;
    const int m  = m0 + t;

    const float qx = xyzout[((size_t)b * kM + m) * 3 + 0];
    const float qy = xyzout[((size_t)b * kM + m) * 3 + 1];
    const float qz = xyzout[((size_t)b * kM + m) * 3 + 2];
    const float so = qx * qx + qy * qy + qz * qz;

    float d0 = 3.4e38f, d1 = 3.4e38f, d2 = 3.4e38f;
    int   i0 = 0,       i1 = 0,       i2 = 0;

    for (int ch = 0; ch < kNIN / 1024; ++ch) {
        const char* src =
            (const char*)(xyzin + ((size_t)b * kNIN + ch * 1024) * 3); // 12288 B
#if __has_builtin(__builtin_amdgcn_global_load_async_to_lds_b128) && \
    __has_builtin(__builtin_amdgcn_s_wait_asynccnt)
        // async DMA: 768 x 16B lane-copies across the block
        #pragma unroll
        for (int j = 0; j < 3; ++j) {
            int chunk = t + 256 * j;
            __builtin_amdgcn_global_load_async_to_lds_b128(
                glb_v4(src + chunk * 16),
                lds_v4((char*)smRaw + chunk * 16), 0, 0);
        }
        __builtin_amdgcn_s_wait_asynccnt(0);
#else
        #pragma unroll
        for (int j = 0; j < 3; ++j) {
            float4 vsrc = *(const float4*)(src + (t + 256 * j) * 16);
            *((float4*)smRaw + t + 256 * j) = vsrc;
        }
#endif
        __syncthreads();

        // append |p|^2 into broadcast table
        #pragma unroll
        for (int j = 0; j < 4; ++j) {
            int p = (t << 2) + j;
            float px = smRaw[p * 3 + 0];
            float py = smRaw[p * 3 + 1];
            float pz = smRaw[p * 3 + 2];
            smPts[p] = make_float4(px, py, pz, px * px + py * py + pz * pz);
        }
        __syncthreads();

        #pragma unroll 4
        for (int p = 0; p < 1024; ++p) {       // uniform -> LDS broadcast
            float4 q = smPts[p];
            float d  = so + q.w;
            d = fmaf(-2.0f * qx, q.x, d);
            d = fmaf(-2.0f * qy, q.y, d);
            d = fmaf(-2.0f * qz, q.z, d);
            if (d < 0.0f) d = 1e-7f;           // reference clamp
            int n = ch * 1024 + p;
            if (d < d2) {
                if (d < d1) {
                    if (d < d0) { d2 = d1; i2 = i1; d1 = d0; i1 = i0; d0 = d; i0 = n; }
                    else        { d2 = d1; i2 = i1; d1 = d;  i1 = n; }
                } else          { d2 = d;  i2 = n; }
            }
        }
        __syncthreads();
    }

    // normalized inverse-distance weights
    {
        float a0 = 1.0f / d0, a1 = 1.0f / d1, a2 = 1.0f / d2;
        float s  = 1.0f / (a0 + a1 + a2);
        smIdx[t * 3 + 0] = i0; smIdx[t * 3 + 1] = i1; smIdx[t * 3 + 2] = i2;
        smW[t * 3 + 0] = a0 * s; smW[t * 3 + 1] = a1 * s; smW[t * 3 + 2] = a2 * s;
    }
    __syncthreads();

    // interpolation: 64 m's per sub-pass; gather (coalesced float4 rows of
    // featT) -> LDS transposed -> coalesced store along M.
    for (int sub = 0; sub < 4; ++sub) {
        const int r    = t >> 2;               // 0..63 local m
        const int ml   = (sub << 6) + r;       // 0..255
        const int cseg = (t & 3) << 5;         // 0,32,64,96

        int   j0 = smIdx[ml * 3 + 0], j1 = smIdx[ml * 3 + 1], j2 = smIdx[ml * 3 + 2];
        float a0 = smW[ml * 3 + 0],   a1 = smW[ml * 3 + 1],   a2 = smW[ml * 3 + 2];
        const float* f0 = featT + ((size_t)b * kNIN + j0) * kCO;
        const float* f1 = featT + ((size_t)b * kNIN + j1) * kCO;
        const float* f2 = featT + ((size_t)b * kNIN + j2) * kCO;

        #pragma unroll
        for (int c = 0; c < 32; c += 4) {
            int cc = cseg + c;
            float4 x0 = *(const float4*)(f0 + cc);
            float4 x1 = *(const float4*)(f1 + cc);
            float4 x2 = *(const float4*)(f2 + cc);
            smOutT[(cc + 0) * 65 + r] = a0 * x0.x + a1 * x1.x + a2 * x2.x;
            smOutT[(cc + 1) * 65 + r] = a0 * x0.y + a1 * x1.y + a2 * x2.y;
            smOutT[(cc + 2) * 65 + r] = a0 * x0.z + a1 * x1.z + a2 * x2.z;
            smOutT[(cc + 3) * 65 + r] = a0 * x0.w + a1 * x1.w + a2 * x2.w;
        }
        __syncthreads();

        const int mw = t & 63;
        #pragma unroll 8
        for (int co = 0; co < 32; ++co) {
            int c = (co << 2) + (t >> 6);
            out[((size_t)b * kCO + c) * kM + m0 + (sub << 6) + mw] = smOutT[c * 65 + mw];
        }
        __syncthreads();
    }
}

// ---------------- launch ----------------
extern "C" void kernel_launch(void* const* d_in, const int* in_sizes, int n_in,
                              void* d_out, int out_size, void* d_ws, size_t ws_size,
                              hipStream_t stream) {
    const float* rgb    = (const float*)d_in[0];
    const float* xyzin  = (const float*)d_in[1];
    const float* xyzout = (const float*)d_in[2];
    const float* w1  = (const float*)d_in[3];
    const float* b1  = (const float*)d_in[4];
    const float* g1  = (const float*)d_in[5];
    const float* be1 = (const float*)d_in[6];
    const float* rm1 = (const float*)d_in[7];
    const float* rv1 = (const float*)d_in[8];
    const float* w2  = (const float*)d_in[9];
    const float* b2  = (const float*)d_in[10];
    const float* g2  = (const float*)d_in[11];
    const float* be2 = (const float*)d_in[12];
    const float* rm2 = (const float*)d_in[13];
    const float* rv2 = (const float*)d_in[14];
    const float* w3  = (const float*)d_in[15];
    const float* b3  = (const float*)d_in[16];
    const float* g3  = (const float*)d_in[17];
    const float* be3 = (const float*)d_in[18];
    const float* rm3 = (const float*)d_in[19];
    const float* rv3 = (const float*)d_in[20];

    float* featT = (float*)d_ws;                               // B*NIN*CO floats (8 MB)
    float* pars  = featT + (size_t)kB * kNIN * kCO;            // 768 floats

    fold_bn_kernel<<<1, 128, 0, stream>>>(b1, g1, be1, rm1, rv1,
                                          b2, g2, be2, rm2, rv2,
                                          b3, g3, be3, rm3, rv3, pars);

    mlp_fused_kernel<<<kB * (kNIN / 16), 256, 0, stream>>>(rgb, w1, w2, w3, pars, featT);

    interp_kernel<<<kB * (kM / 256), 256, 0, stream>>>(xyzin, xyzout, featT, (float*)d_out);
}